// WaveletConvolution_50955491999769
// MI455X (gfx1250) — compile-verified
//
#include <hip/hip_runtime.h>

// Problem constants (fixed by the reference).
constexpr int D_FEAT = 256;   // K
constexpr int UNITS  = 128;   // N of the dense projection / feature width of h

typedef __attribute__((ext_vector_type(2))) float v2f;
typedef __attribute__((ext_vector_type(8))) float v8f;

// ---------------------------------------------------------------------------
// Dense projection h = x @ kernel using V_WMMA_F32_16X16X4_F32 (exact fp32).
// One wave handles a 16-row band of x and ALL 128 output columns:
//   8 accumulator tiles (16x16 each), A fragment loaded once per k-step,
//   so x is read exactly once from HBM; kernel (128 KB) stays in L1/L2.
// Layouts per CDNA5 ISA 7.12.2 (32-bit A 16x4 / 32-bit C 16x16).
// ---------------------------------------------------------------------------
__global__ __launch_bounds__(256) void wavelet_gemm_wmma(
    const float* __restrict__ x, const float* __restrict__ w,
    float* __restrict__ h, int n_nodes, int m_tiles) {
  const int wave = (int)((blockIdx.x * blockDim.x + threadIdx.x) >> 5);
  const int lane = (int)(threadIdx.x & 31);
  if (wave >= m_tiles) return;                 // wave-uniform exit (EXEC stays full)

  const int m0    = wave * 16;
  const int lmod  = lane & 15;
  const int khalf = (lane >> 4) << 1;          // 0 for lanes 0-15, 2 for lanes 16-31

  int mrow = m0 + lmod;
  if (mrow >= n_nodes) mrow = n_nodes - 1;     // clamp (keeps EXEC all-1 for WMMA)
  const float* xrow = x + (size_t)mrow * D_FEAT + khalf;

  v8f acc[8] = {};                             // 8 tiles x 8 VGPRs = full N=128

#pragma unroll 4
  for (int kk = 0; kk < D_FEAT; kk += 4) {
    // A fragment: lane l -> {x[m][kk+khalf], x[m][kk+khalf+1]} (8B aligned)
    v2f a = *(const v2f*)(xrow + kk);
    const float* brow = w + (size_t)(kk + khalf) * UNITS + lmod;
#pragma unroll
    for (int nt = 0; nt < 8; ++nt) {
      // B fragment: lane l -> {w[kk+khalf][n], w[kk+khalf+1][n]}, n = nt*16 + (l&15)
      v2f b;
      b.x = brow[nt * 16];
      b.y = brow[nt * 16 + UNITS];
      acc[nt] = __builtin_amdgcn_wmma_f32_16x16x4_f32(
          false, a, false, b, (short)0, acc[nt], false, false);
    }
  }

  // C/D layout: acc[v] holds (M = m0 + v + (lane<16?0:8), N = nt*16 + (l&15))
  const int mstore = m0 + ((lane >> 4) << 3);
#pragma unroll
  for (int v = 0; v < 8; ++v) {
    const int row = mstore + v;
    if (row < n_nodes) {
      float* hrow = h + (size_t)row * UNITS + lmod;
#pragma unroll
      for (int nt = 0; nt < 8; ++nt) hrow[nt * 16] = acc[nt][v];
    }
  }
}

// ---------------------------------------------------------------------------
// Edge-parallel SpMM scatter: out[rows[e]] += vals[e] * (filt?[cols[e]]) * h[cols[e]]
// One wave per edge: 32 lanes x float4 = the full 128-float row.
// Working set (h/out ~51 MB each) is L2-resident on MI455X (192 MB L2),
// so gathers + float atomics are serviced at L2 bandwidth.
// ---------------------------------------------------------------------------
__global__ __launch_bounds__(256) void spmm_scatter(
    const int* __restrict__ rows, const int* __restrict__ cols,
    const float* __restrict__ vals, const float* __restrict__ filt,
    const float* __restrict__ h, float* __restrict__ out, int n_edges) {
  const unsigned gid = blockIdx.x * blockDim.x + threadIdx.x;
  const int e    = (int)(gid >> 5);
  const int lane = (int)(gid & 31);
  if (e >= n_edges) return;

  const int r = rows[e];
  const int c = cols[e];
  float v = vals[e];
  if (filt) v *= filt[c];                      // fused diagonal filter (row scale of h2)

  const float4 hv = ((const float4*)(h + (size_t)c * UNITS))[lane];
  float* o = out + (size_t)r * UNITS + lane * 4;
  atomicAdd(o + 0, v * hv.x);
  atomicAdd(o + 1, v * hv.y);
  atomicAdd(o + 2, v * hv.z);
  atomicAdd(o + 3, v * hv.w);
}

__global__ __launch_bounds__(256) void zero_f4(float4* __restrict__ p, long n4) {
  const long i = (long)blockIdx.x * blockDim.x + threadIdx.x;
  if (i < n4) p[i] = make_float4(0.f, 0.f, 0.f, 0.f);
}

// ---------------------------------------------------------------------------
extern "C" void kernel_launch(void* const* d_in, const int* in_sizes, int n_in,
                              void* d_out, int out_size, void* d_ws, size_t ws_size,
                              hipStream_t stream) {
  const float* x       = (const float*)d_in[0];
  const float* kern    = (const float*)d_in[1];
  const float* filt    = (const float*)d_in[2];
  const float* w_vals  = (const float*)d_in[3];
  const float* iw_vals = (const float*)d_in[4];
  const int*   w_rows  = (const int*)d_in[5];
  const int*   w_cols  = (const int*)d_in[6];
  const int*   iw_rows = (const int*)d_in[7];
  const int*   iw_cols = (const int*)d_in[8];

  const int n_nodes = in_sizes[2];   // filt length == N_NODES
  const int n_w     = in_sizes[3];   // wavelet edges
  const int n_iw    = in_sizes[4];   // inverse-wavelet edges

  const size_t hfloats = (size_t)n_nodes * UNITS;
  const size_t hbytes  = hfloats * sizeof(float);
  const long   n4      = (long)(hfloats / 4);

  float* hbuf = (float*)d_ws;

  const dim3 blk(256);
  const int  m_tiles = (n_nodes + 15) / 16;
  const dim3 gemm_grid((unsigned)(((long)m_tiles * 32 + 255) / 256));
  const dim3 zgrid((unsigned)((n4 + 255) / 256));

  // Stage 1: h = x @ kernel (WMMA fp32)
  wavelet_gemm_wmma<<<gemm_grid, blk, 0, stream>>>(x, kern, hbuf, n_nodes, m_tiles);

  auto spmm_grid = [&](int ne) {
    return dim3((unsigned)(((long)ne * 32 + 255) / 256));
  };

  if (ws_size >= 2 * hbytes) {
    // Plan A: both intermediates live in workspace.
    float* h2 = (float*)((char*)d_ws + hbytes);
    zero_f4<<<zgrid, blk, 0, stream>>>((float4*)h2, n4);
    spmm_scatter<<<spmm_grid(n_iw), blk, 0, stream>>>(
        iw_rows, iw_cols, iw_vals, nullptr, hbuf, h2, n_iw);
    zero_f4<<<zgrid, blk, 0, stream>>>((float4*)d_out, n4);
    spmm_scatter<<<spmm_grid(n_w), blk, 0, stream>>>(
        w_rows, w_cols, w_vals, filt, h2, (float*)d_out, n_w);
  } else {
    // Plan B: single ws buffer; bounce through d_out, fix up with a D2D copy.
    zero_f4<<<zgrid, blk, 0, stream>>>((float4*)d_out, n4);
    spmm_scatter<<<spmm_grid(n_iw), blk, 0, stream>>>(
        iw_rows, iw_cols, iw_vals, nullptr, hbuf, (float*)d_out, n_iw);  // d_out = h2
    zero_f4<<<zgrid, blk, 0, stream>>>((float4*)hbuf, n4);
    spmm_scatter<<<spmm_grid(n_w), blk, 0, stream>>>(
        w_rows, w_cols, w_vals, filt, (const float*)d_out, hbuf, n_w);
    hipMemcpyAsync(d_out, hbuf, hbytes, hipMemcpyDeviceToDevice, stream);
  }
}